// SmartDerivatives_25683904430509
// MI455X (gfx1250) — compile-verified
//
#include <hip/hip_runtime.h>
#include <stddef.h>

#define N_ATOMS   100
#define DESC      4950            // N*(N-1)/2
#define OUTPF     (N_ATOMS * 3)   // 300 outputs per frame
#define TPB       256
#define NWAVES    (TPB / 32)
#define REPSTRIDE 304             // 300 padded to spread LDS banks
#define PACK      4               // descriptors per lane per iteration

#if defined(__gfx1250__) && __has_builtin(__builtin_amdgcn_global_load_async_to_lds_b32)
#define HAVE_ASYNC_LDS 1
#else
#define HAVE_ASYNC_LDS 0
#endif

typedef int __attribute__((address_space(1))) gint_as1;
typedef int __attribute__((address_space(3))) lint_as3;

// row start of triu row i (N=100): S(i) = i*(199-i)/2
__device__ __forceinline__ int row_start(int i) { return (i * (199 - i)) >> 1; }

// decode triu pair index d -> (i, j), row-major np.triu_indices(100, 1) order
__device__ __forceinline__ void decode_pair(int d, int& i_out, int& j_out) {
  float fd = (float)d;
  int i = (int)((199.0f - __builtin_sqrtf(39601.0f - 8.0f * fd)) * 0.5f);
  i = i < 0 ? 0 : (i > 98 ? 98 : i);
  while (i < 98 && row_start(i + 1) <= d) ++i;   // float-precision fixup
  while (i > 0 && row_start(i) > d) --i;
  i_out = i;
  j_out = d - row_start(i) + i + 1;
}

__global__ __launch_bounds__(TPB) void smartderiv_kernel(
    const float* __restrict__ x,     // [BATCH, DESC]
    const float* __restrict__ left,  // [BATCH, DESC, 6]
    float* __restrict__ out) {       // [BATCH, 300]
  __shared__ __align__(16) float sx[DESC];      // staged x row (19.8 KB)
  __shared__ float srep[NWAVES * REPSTRIDE];    // per-wave accumulator replicas

  const int b    = blockIdx.x;
  const int tid  = threadIdx.x;
  const int lane = tid & 31;
  float* myrep = &srep[(tid >> 5) * REPSTRIDE];

  for (int k = tid; k < NWAVES * REPSTRIDE; k += TPB) srep[k] = 0.0f;

  const float* xrow = x + (size_t)b * DESC;
#if HAVE_ASYNC_LDS
  {
    gint_as1* gp = (gint_as1*)xrow;
    lint_as3* lp = (lint_as3*)sx;
    for (int d = tid; d < DESC; d += TPB)
      __builtin_amdgcn_global_load_async_to_lds_b32(gp + d, lp + d, 0, 0);
#if __has_builtin(__builtin_amdgcn_s_wait_asynccnt)
    __builtin_amdgcn_s_wait_asynccnt(0);
#else
    asm volatile("s_wait_asynccnt 0" ::: "memory");
#endif
  }
#else
  for (int d = tid; d < DESC; d += TPB) sx[d] = xrow[d];
#endif
  __syncthreads();

  const float* lbase = left + (size_t)b * DESC * 6;

  for (int base = 0; base < DESC; base += TPB * PACK) {
    const int d0 = base + tid * PACK;

    if (base + TPB * PACK < DESC)  // stream-ahead on the big left[] array
      __builtin_prefetch(lbase + (size_t)(d0 + TPB * PACK) * 6, 0, 1);

    int   i0 = -1;                      // segment key: row of first pack element
    float a0 = 0.f, a1 = 0.f, a2 = 0.f; // i0-row accumulators

    if (d0 < DESC) {
      int ic, jc;
      decode_pair(d0, ic, jc);
      i0 = ic;

      float xq[PACK];
      float lv[PACK][6];
      if (d0 + PACK <= DESC) {
        // fast path: 16B-aligned vector loads (d0 multiple of 4)
        float4 xv4 = *(const float4*)&sx[d0];
        xq[0] = xv4.x; xq[1] = xv4.y; xq[2] = xv4.z; xq[3] = xv4.w;
        const float4* lp = (const float4*)(lbase + (size_t)d0 * 6);
        float4 v0 = lp[0], v1 = lp[1], v2 = lp[2], v3 = lp[3], v4 = lp[4], v5 = lp[5];
        lv[0][0]=v0.x; lv[0][1]=v0.y; lv[0][2]=v0.z; lv[0][3]=v0.w; lv[0][4]=v1.x; lv[0][5]=v1.y;
        lv[1][0]=v1.z; lv[1][1]=v1.w; lv[1][2]=v2.x; lv[1][3]=v2.y; lv[1][4]=v2.z; lv[1][5]=v2.w;
        lv[2][0]=v3.x; lv[2][1]=v3.y; lv[2][2]=v3.z; lv[2][3]=v3.w; lv[2][4]=v4.x; lv[2][5]=v4.y;
        lv[3][0]=v4.z; lv[3][1]=v4.w; lv[3][2]=v5.x; lv[3][3]=v5.y; lv[3][4]=v5.z; lv[3][5]=v5.w;
      } else {
        // tail: guarded scalar loads, never read past the allocations
        #pragma unroll
        for (int q = 0; q < PACK; ++q) {
          bool v = (d0 + q) < DESC;
          xq[q] = v ? sx[d0 + q] : 0.f;
          #pragma unroll
          for (int k = 0; k < 6; ++k)
            lv[q][k] = v ? lbase[(size_t)(d0 + q) * 6 + k] : 0.f;
        }
      }

      #pragma unroll
      for (int q = 0; q < PACK; ++q) {
        if ((d0 + q) < DESC) {
          float xv = xq[q];
          float p0 = lv[q][0] * xv, p1 = lv[q][1] * xv, p2 = lv[q][2] * xv;
          // j-side: distinct addresses per lane -> direct ds_add_f32
          atomicAdd(&myrep[3 * jc + 0], lv[q][3] * xv);
          atomicAdd(&myrep[3 * jc + 1], lv[q][4] * xv);
          atomicAdd(&myrep[3 * jc + 2], lv[q][5] * xv);
          if (ic == i0) { a0 += p0; a1 += p1; a2 += p2; }
          else {        // pack crossed a row boundary (rare): direct atomics
            atomicAdd(&myrep[3 * ic + 0], p0);
            atomicAdd(&myrep[3 * ic + 1], p1);
            atomicAdd(&myrep[3 * ic + 2], p2);
          }
        }
        // incremental pair advance: (i, j) -> next triu pair
        if (++jc >= N_ATOMS) { ++ic; jc = ic + 1; }
      }
    }

    // i0 is non-decreasing across lanes: segmented suffix reduction over
    // equal-i0 runs; only run leaders touch LDS. All lanes participate.
#pragma unroll
    for (int off = 1; off < 32; off <<= 1) {
      int   oi = __shfl_down(i0, off, 32);
      float b0 = __shfl_down(a0, off, 32);
      float b1 = __shfl_down(a1, off, 32);
      float b2 = __shfl_down(a2, off, 32);
      if ((lane + off) < 32 && oi == i0) { a0 += b0; a1 += b1; a2 += b2; }
    }
    int pi = __shfl_up(i0, 1, 32);
    bool leader = (lane == 0) || (pi != i0);
    if (leader && i0 >= 0) {
      atomicAdd(&myrep[3 * i0 + 0], a0);
      atomicAdd(&myrep[3 * i0 + 1], a1);
      atomicAdd(&myrep[3 * i0 + 2], a2);
    }
  }
  __syncthreads();

  for (int m = tid; m < OUTPF; m += TPB) {
    float s = 0.f;
#pragma unroll
    for (int r = 0; r < NWAVES; ++r) s += srep[r * REPSTRIDE + m];
    out[(size_t)b * OUTPF + m] = s * s;
  }
}

extern "C" void kernel_launch(void* const* d_in, const int* in_sizes, int n_in,
                              void* d_out, int out_size, void* d_ws, size_t ws_size,
                              hipStream_t stream) {
  // inputs: x [B*D] f32, left [nnz] f32, batch_ind, des_ind, scatter_idx (unused:
  // the index structure is closed-form and recomputed on device).
  const float* x    = (const float*)d_in[0];
  const float* left = (const float*)d_in[1];
  float* out = (float*)d_out;
  (void)in_sizes; (void)n_in; (void)d_ws; (void)ws_size;

  int batch = out_size / OUTPF;  // 1024
  smartderiv_kernel<<<batch, TPB, 0, stream>>>(x, left, out);
}